// AmberDynamics_46660524704099
// MI455X (gfx1250) — compile-verified
//
#include <hip/hip_runtime.h>

// ---------------------------------------------------------------------------
// AmberDynamics: out[0:5]=qdot ; out[5:10]=D^-1 (Bu - H)
// D = 2I + 0.3(c c^T + s s^T)  (c=cos q, s=sin q)  -> Woodbury rank-2:
//   D^-1 r = r/2 - (1/4) U (M^-1) U^T r,  U=[c s],  M = (10/3)I2 + (1/2)U^T U
// Memory-bound: 96 B/elem, ~96 MB total -> ~4.1 us at 23.3 TB/s.
// gfx1250 path: async global->LDS staging of the 40B-stride AoS state
// (global_load_async_to_lds_b32), s_wait_asynccnt fence, pitch-11 LDS reads.
// ---------------------------------------------------------------------------

#define BLOCK 256
#define PITCH 11   // 10 floats/elem padded to 11 -> gcd(11,64)=1, no bank conflicts

typedef __attribute__((address_space(3))) float lds_f32;

__device__ __forceinline__ void async_ld_b32(unsigned ldsaddr, unsigned long long gaddr)
{
    asm volatile("global_load_async_to_lds_b32 %0, %1, off"
                 :: "v"(ldsaddr), "v"(gaddr) : "memory");
}

__global__ __launch_bounds__(BLOCK)
void amber_dyn_kernel(const float* __restrict__ state,
                      const float* __restrict__ u,
                      float* __restrict__ out,
                      int B)
{
    __shared__ float s_state[BLOCK * PITCH];

    const int tid = threadIdx.x;
    const long long elemBase = (long long)blockIdx.x * BLOCK;
    const int nElems = (int)(((long long)B - elemBase) < BLOCK ? ((long long)B - elemBase)
                                                               : (long long)BLOCK);
    const int nDw = nElems * 10;

    // LDS byte offset of s_state (addrspace(3) pointers are LDS-relative)
    const unsigned ldsBase = (unsigned)(unsigned long long)(lds_f32*)s_state;
    const float* gbase = state + elemBase * 10;

    // Stage this block's state slab (nElems*10 dwords) into LDS, coalesced,
    // de-swizzling 10-float rows onto an 11-float pitch.
    if (nDw == BLOCK * 10) {
        // Uniform fast path: no per-issue divergence guards.
#pragma unroll
        for (int it = 0; it < 10; ++it) {
            const int g = tid + it * BLOCK;
            const int e = g / 10;            // const-div -> mul/shift
            const int k = g - e * 10;
            async_ld_b32(ldsBase + (unsigned)(e * PITCH + k) * 4u,
                         (unsigned long long)(gbase + g));
        }
    } else {
        for (int it = 0; it < 10; ++it) {
            const int g = tid + it * BLOCK;
            if (g < nDw) {
                const int e = g / 10;
                const int k = g - e * 10;
                async_ld_b32(ldsBase + (unsigned)(e * PITCH + k) * 4u,
                             (unsigned long long)(gbase + g));
            }
        }
    }

    // Overlap the u fetch with the async staging (clamped, branch-free).
    const long long elem = elemBase + tid;
    const long long elemC = elem < (long long)B ? elem : (long long)B - 1;
    const float4 uv = *(const float4*)(u + elemC * 4);   // 16B aligned

    asm volatile("s_wait_asynccnt 0" ::: "memory");
    __syncthreads();

    if (elem >= (long long)B) return;   // no barriers past this point

    const float* me = &s_state[tid * PITCH];
    float q[5], qd[5], s[5], c[5];
#pragma unroll
    for (int j = 0; j < 5; ++j) { q[j] = me[j]; qd[j] = me[5 + j]; }

    float v2 = 0.f;
#pragma unroll
    for (int j = 0; j < 5; ++j) v2 = fmaf(qd[j], qd[j], v2);

#pragma unroll
    for (int j = 0; j < 5; ++j) __sincosf(q[j], &s[j], &c[j]);

    // rhs r = B*u - H ; B*u = [0,u0,u1,u2,u3]
    const float coef = fmaf(0.1f, v2, 9.8f);
    float r[5];
    r[0] = 0.f  - fmaf(coef, s[0], 0.05f * qd[0]);
    r[1] = uv.x - fmaf(coef, s[1], 0.05f * qd[1]);
    r[2] = uv.y - fmaf(coef, s[2], 0.05f * qd[2]);
    r[3] = uv.z - fmaf(coef, s[3], 0.05f * qd[3]);
    r[4] = uv.w - fmaf(coef, s[4], 0.05f * qd[4]);

    // Woodbury: Gram of U=[c s] and projections of r
    float cc = 0.f, cs = 0.f, ss = 0.f, pa = 0.f, pb = 0.f;
#pragma unroll
    for (int j = 0; j < 5; ++j) {
        cc = fmaf(c[j], c[j], cc);
        cs = fmaf(c[j], s[j], cs);
        ss = fmaf(s[j], s[j], ss);
        pa = fmaf(c[j], r[j], pa);
        pb = fmaf(s[j], r[j], pb);
    }
    const float m00 = fmaf(0.5f, cc, 10.0f / 3.0f);
    const float m11 = fmaf(0.5f, ss, 10.0f / 3.0f);
    const float m01 = 0.5f * cs;
    const float invdet = 1.0f / (m00 * m11 - m01 * m01);
    const float y1 = (m11 * pa - m01 * pb) * invdet;
    const float y2 = (m00 * pb - m01 * pa) * invdet;

    float x[5];
#pragma unroll
    for (int j = 0; j < 5; ++j)
        x[j] = fmaf(0.5f, r[j], -0.25f * fmaf(c[j], y1, s[j] * y2));

    // out row is 40B-strided -> only 8B aligned; use float2 stores
    float* op = out + elem * 10;
    ((float2*)op)[0] = make_float2(qd[0], qd[1]);
    ((float2*)op)[1] = make_float2(qd[2], qd[3]);
    ((float2*)op)[2] = make_float2(qd[4], x[0]);
    ((float2*)op)[3] = make_float2(x[1],  x[2]);
    ((float2*)op)[4] = make_float2(x[3],  x[4]);
}

extern "C" void kernel_launch(void* const* d_in, const int* in_sizes, int n_in,
                              void* d_out, int out_size, void* d_ws, size_t ws_size,
                              hipStream_t stream)
{
    // setup_inputs order: t (1), state (B*10), u (B*4)
    const float* state = (const float*)d_in[1];
    const float* u     = (const float*)d_in[2];
    float* out = (float*)d_out;
    const int B = in_sizes[1] / 10;
    const int grid = (B + BLOCK - 1) / BLOCK;
    amber_dyn_kernel<<<grid, BLOCK, 0, stream>>>(state, u, out, B);
}